// CapsuleLinear_6184752906707
// MI455X (gfx1250) — compile-verified
//
#include <hip/hip_runtime.h>
#include <math.h>

typedef __attribute__((ext_vector_type(2))) float v2f;
typedef __attribute__((ext_vector_type(8))) float v8f;

#define NB      32
#define NCAPS   1152
#define ILEN    32
#define OCAPS   64
#define OLEN    32
#define NITER   3
#define THREADS 256
#define NWAVES  (THREADS / 32)
#define XPITCH  33   // pad LDS pitch to avoid bank conflicts on column reads
#define VPITCH  33

__global__ __launch_bounds__(THREADS)
void capsule_routing_kernel(const float* __restrict__ x,
                            const float* __restrict__ weight,
                            float* __restrict__ out,
                            float* __restrict__ logits_ws,
                            float* __restrict__ probs_ws) {
  // LDS: X tile (padded) + S + output + V  ~= 173 KB of the 320 KB/WGP
  __shared__ float XL[NCAPS * XPITCH];
  __shared__ float SL[OCAPS * ILEN];
  __shared__ float OL[OCAPS * OLEN];
  __shared__ float VL[OCAPS * VPITCH];

  const int b    = blockIdx.x;
  const int tid  = threadIdx.x;
  const int lane = tid & 31;      // wave32
  const int wid  = tid >> 5;

  const float* xb = x + (size_t)b * NCAPS * ILEN;
  float* logits   = logits_ws + (size_t)b * OCAPS * NCAPS;
  float* probs    = probs_ws  + (size_t)b * OCAPS * NCAPS;

  // Load X[b] (1152x32 fp32) into LDS once.
  for (int idx = tid; idx < NCAPS * ILEN; idx += THREADS) {
    int n = idx >> 5, i = idx & 31;
    XL[n * XPITCH + i] = xb[idx];
  }
  __syncthreads();

  for (int it = 0; it < NITER; ++it) {
    // ---- Phase P: probs = softmax over o of logits (per n, thread-local) ----
    if (it == 0) {
      const float u = 1.0f / (float)OCAPS;   // logits==0 -> uniform coupling
      for (int idx = tid; idx < OCAPS * NCAPS; idx += THREADS) probs[idx] = u;
    } else {
      for (int n = tid; n < NCAPS; n += THREADS) {
        float lv[OCAPS];
        float m = -1e30f;
        #pragma unroll
        for (int o = 0; o < OCAPS; ++o) {
          lv[o] = logits[o * NCAPS + n];
          m = fmaxf(m, lv[o]);
        }
        float d = 0.0f;
        #pragma unroll
        for (int o = 0; o < OCAPS; ++o) { lv[o] = __expf(lv[o] - m); d += lv[o]; }
        float rd = 1.0f / d;
        #pragma unroll
        for (int o = 0; o < OCAPS; ++o) probs[o * NCAPS + n] = lv[o] * rd;
      }
    }
    __threadfence();
    __syncthreads();

    // ---- Phase S (WMMA f32 16x16x4): S[o,i] = sum_n probs[o,n] * X[n,i] ----
    // 64x32 output = 4x2 tiles, one tile per wave; K = 1152.
    {
      const int mT = wid >> 1;           // o-tile 0..3
      const int nT = wid & 1;            // i-tile 0..1
      const int m  = lane & 15;
      const int kh = lane >> 4;          // which K-half this lane holds
      const int nn = nT * 16 + m;
      v8f c = {0.f,0.f,0.f,0.f,0.f,0.f,0.f,0.f};
      const float2* prow =
          (const float2*)(probs + (size_t)(mT * 16 + m) * NCAPS);
      #pragma unroll 4
      for (int k = 0; k < NCAPS; k += 4) {
        float2 av = prow[(k >> 1) + kh];               // A: probs[M, k+2kh .. +1]
        v2f a; a.x = av.x; a.y = av.y;
        v2f bb;                                        // B: X[k+2kh, nn]
        bb.x = XL[(k + 2 * kh)     * XPITCH + nn];
        bb.y = XL[(k + 2 * kh + 1) * XPITCH + nn];
        c = __builtin_amdgcn_wmma_f32_16x16x4_f32(
                false, a, false, bb, (short)0, c, false, false);
      }
      #pragma unroll
      for (int r8 = 0; r8 < 8; ++r8) {
        int mm = mT * 16 + kh * 8 + r8;                // D layout: M = 8*kh + vgpr
        SL[mm * ILEN + nn] = c[r8];
      }
    }
    __syncthreads();

    // ---- Phase O: output[o,l] = S[o,:] . W[o,l,:]; then squash ----
    for (int idx = tid; idx < OCAPS * OLEN; idx += THREADS) {
      int o = idx >> 5, l = idx & 31;
      const float* wr = weight + (size_t)(o * OLEN + l) * ILEN;
      const float* sr = SL + o * ILEN;
      float acc = 0.0f;
      #pragma unroll
      for (int i = 0; i < ILEN; ++i) acc = fmaf(sr[i], wr[i], acc);
      OL[idx] = acc;
    }
    __syncthreads();
    if (tid < OCAPS) {
      float ns = 0.0f;
      #pragma unroll
      for (int l = 0; l < OLEN; ++l) { float t = OL[tid * OLEN + l]; ns += t * t; }
      float scale = sqrtf(ns) / (1.0f + ns);   // v*||v||/(1+||v||^2)
      #pragma unroll
      for (int l = 0; l < OLEN; ++l) OL[tid * OLEN + l] *= scale;
    }
    __syncthreads();

    if (it == NITER - 1) {
      float* ob = out + (size_t)b * OCAPS * OLEN;
      for (int idx = tid; idx < OCAPS * OLEN; idx += THREADS) ob[idx] = OL[idx];
      break;
    }

    // ---- Phase V: v[o,i] = sum_l W[o,l,i] * uhat[o,l] ----
    for (int idx = tid; idx < OCAPS * ILEN; idx += THREADS) {
      int o = idx >> 5, i = idx & 31;
      const float* wr  = weight + (size_t)o * OLEN * ILEN + i;
      const float* orow = OL + o * OLEN;
      float acc = 0.0f;
      #pragma unroll
      for (int l = 0; l < OLEN; ++l) acc = fmaf(wr[l * ILEN], orow[l], acc);
      VL[o * VPITCH + i] = acc;
    }
    __syncthreads();

    // ---- Phase D (WMMA f32 16x16x4): delta[n,o] = X[n,:] . v[o,:]; logits (+)= delta
    // 1152x64 output = 72x4 = 288 tiles over 8 waves; K = 32.
    for (int t = wid; t < (NCAPS / 16) * (OCAPS / 16); t += NWAVES) {
      const int mT = t >> 2;            // n-tile 0..71
      const int nT = t & 3;             // o-tile 0..3
      const int m  = lane & 15;
      const int kh = lane >> 4;
      const int nrow = mT * 16 + m;     // A row (n)
      const int ocol = nT * 16 + m;     // B col (o)
      v8f c = {0.f,0.f,0.f,0.f,0.f,0.f,0.f,0.f};
      #pragma unroll
      for (int k = 0; k < ILEN; k += 4) {
        v2f a, bb;
        a.x  = XL[nrow * XPITCH + k + 2 * kh];
        a.y  = XL[nrow * XPITCH + k + 2 * kh + 1];
        bb.x = VL[ocol * VPITCH + k + 2 * kh];
        bb.y = VL[ocol * VPITCH + k + 2 * kh + 1];
        c = __builtin_amdgcn_wmma_f32_16x16x4_f32(
                false, a, false, bb, (short)0, c, false, false);
      }
      #pragma unroll
      for (int r8 = 0; r8 < 8; ++r8) {
        int n = mT * 16 + kh * 8 + r8;
        int o = nT * 16 + m;
        float* lp = logits + (size_t)o * NCAPS + n;
        if (it == 0) *lp = c[r8];       // first write: no zero-init of ws needed
        else         *lp += c[r8];
      }
    }
    __threadfence();
    __syncthreads();
  }
}

extern "C" void kernel_launch(void* const* d_in, const int* in_sizes, int n_in,
                              void* d_out, int out_size, void* d_ws, size_t ws_size,
                              hipStream_t stream) {
  (void)in_sizes; (void)n_in; (void)out_size; (void)ws_size;
  const float* x = (const float*)d_in[0];      // [32,1152,32] fp32
  const float* w = (const float*)d_in[1];      // [64,32,32]  fp32
  float* out     = (float*)d_out;              // [32,64,32]  fp32
  // workspace: logits + probs, each 32*64*1152 fp32 (~9.4 MB each)
  float* logits  = (float*)d_ws;
  float* probs   = logits + (size_t)NB * OCAPS * NCAPS;
  capsule_routing_kernel<<<NB, THREADS, 0, stream>>>(x, w, out, logits, probs);
}